// SSIMLoss_20375324852448
// MI455X (gfx1250) — compile-verified
//
#include <hip/hip_runtime.h>

// ---------------------------------------------------------------------------
// SSIM (global-statistics variant) for 96 channels of 512x512 fp32.
// Memory-bound: ~201 MB read -> ~8.6 us floor at 23.3 TB/s HBM.
// Pass 1: streaming nontemporal b128 loads, accumulation via
//         V_WMMA_F32_16X16X4_F32 (B = ones => D accumulates row sums;
//         exact /16 recovers totals), LDS tree reduce -> 5 partial sums
//         per (channel, segment).
// Pass 2: fold segments, apply SSIM formula, write 96 floats.
// ---------------------------------------------------------------------------

typedef __attribute__((ext_vector_type(2))) float v2f;
typedef __attribute__((ext_vector_type(4))) float v4f;
typedef __attribute__((ext_vector_type(8))) float v8f;

#define SEG 16      // blocks per channel
#define TPB 256     // threads per block (8 waves on wave32)
#define SSIM_C1 0.0001f
#define SSIM_C2 0.001f

__device__ __forceinline__ v8f wmma_acc(v2f a, v8f c) {
    const v2f ones = {1.0f, 1.0f};
    // D(16x16,f32) += A(16x4,f32) x Ones(4x16)  => per-row sums, replicated
    return __builtin_amdgcn_wmma_f32_16x16x4_f32(
        /*neg_a=*/false, a, /*neg_b=*/false, ones,
        /*c_mod=*/(short)0, c, /*reuse_a=*/false, /*reuse_b=*/false);
}

__device__ __forceinline__ float fold8(v8f v) {
    return ((v[0] + v[1]) + (v[2] + v[3])) + ((v[4] + v[5]) + (v[6] + v[7]));
}

__global__ __launch_bounds__(TPB) void ssim_partial_kernel(
    const float* __restrict__ I1, const float* __restrict__ I2,
    float* __restrict__ ws, int nvec /* float4 per channel */,
    int iters /* nvec / (SEG*TPB), uniform */) {

    const int seg = blockIdx.x;   // [0, SEG)
    const int ch  = blockIdx.y;   // [0, channels)
    const int tid = threadIdx.x;

    const long long chanBase = (long long)ch * (long long)nvec;  // v4f units
    const v4f* __restrict__ xp = (const v4f*)I1 + chanBase;
    const v4f* __restrict__ yp = (const v4f*)I2 + chanBase;

    v8f ax = {}, ay = {}, axx = {}, ayy = {}, axy = {};

    // ---- main uniform loop: EXEC all-ones (required for WMMA) ----
    int idx = seg * (TPB * iters) + tid;
    for (int it = 0; it < iters; ++it, idx += TPB) {
        v4f x = __builtin_nontemporal_load(&xp[idx]);   // global_load_b128, TH=NT
        v4f y = __builtin_nontemporal_load(&yp[idx]);

        v2f xlo = {x.x, x.y}, xhi = {x.z, x.w};
        v2f ylo = {y.x, y.y}, yhi = {y.z, y.w};

        ax  = wmma_acc(xlo,       ax);
        ax  = wmma_acc(xhi,       ax);
        ay  = wmma_acc(ylo,       ay);
        ay  = wmma_acc(yhi,       ay);
        axx = wmma_acc(xlo * xlo, axx);
        axx = wmma_acc(xhi * xhi, axx);
        ayy = wmma_acc(ylo * ylo, ayy);
        ayy = wmma_acc(yhi * yhi, ayy);
        axy = wmma_acc(xlo * ylo, axy);
        axy = wmma_acc(xhi * yhi, axy);
    }

    // ---- generic VALU tail (empty for 512x512; seg 0 only) ----
    float tx = 0.f, ty = 0.f, txx = 0.f, tyy = 0.f, txy = 0.f;
    const int tail_start = SEG * TPB * iters;
    if (seg == 0) {
        for (int i = tail_start + tid; i < nvec; i += TPB) {
            v4f x = xp[i];
            v4f y = yp[i];
            tx  += (x.x + x.y) + (x.z + x.w);
            ty  += (y.x + y.y) + (y.z + y.w);
            txx += (x.x * x.x + x.y * x.y) + (x.z * x.z + x.w * x.w);
            tyy += (y.x * y.x + y.y * y.y) + (y.z * y.z + y.w * y.w);
            txy += (x.x * y.x + x.y * y.y) + (x.z * y.z + x.w * y.w);
        }
    }

    // Each WMMA D element is a row-sum replicated across 16 columns:
    // summing all 32 lanes' fold8() gives 16x the true total -> exact /16.
    float sx  = fold8(ax)  * 0.0625f + tx;
    float sy  = fold8(ay)  * 0.0625f + ty;
    float sxx = fold8(axx) * 0.0625f + txx;
    float syy = fold8(ayy) * 0.0625f + tyy;
    float sxy = fold8(axy) * 0.0625f + txy;

    // ---- block LDS tree reduction (stride 5 -> conflict-free, gcd(5,64)=1) ----
    __shared__ float red[TPB * 5];
    red[tid * 5 + 0] = sx;
    red[tid * 5 + 1] = sy;
    red[tid * 5 + 2] = sxx;
    red[tid * 5 + 3] = syy;
    red[tid * 5 + 4] = sxy;
    __syncthreads();
    for (int st = TPB / 2; st > 0; st >>= 1) {
        if (tid < st) {
#pragma unroll
            for (int j = 0; j < 5; ++j)
                red[tid * 5 + j] += red[(tid + st) * 5 + j];
        }
        __syncthreads();
    }

    if (tid == 0) {
        float* o = ws + ((size_t)ch * SEG + seg) * 5;
#pragma unroll
        for (int j = 0; j < 5; ++j) o[j] = red[j];
    }
}

__global__ void ssim_final_kernel(const float* __restrict__ ws,
                                  float* __restrict__ out,
                                  int channels, float n) {
    int c = blockIdx.x * blockDim.x + threadIdx.x;
    if (c >= channels) return;

    float sx = 0.f, sy = 0.f, sxx = 0.f, syy = 0.f, sxy = 0.f;
#pragma unroll
    for (int s = 0; s < SEG; ++s) {
        const float* p = ws + ((size_t)c * SEG + s) * 5;
        sx  += p[0];
        sy  += p[1];
        sxx += p[2];
        syy += p[3];
        sxy += p[4];
    }

    float inv_n = 1.0f / n;
    float mu1 = sx * inv_n;
    float mu2 = sy * inv_n;
    float sig1  = (sxx - sx * sx * inv_n) / (n - 1.0f);   // sum(dx*dx)/(n-1)
    float sig2  = (syy - sy * sy * inv_n) / (n - 1.0f);   // sum(dy*dy)/(n-1)
    float sig12 = (sxy - sx * sy * inv_n) * inv_n;        // mean(dx*dy)

    float num = (2.0f * mu1 * mu2 + SSIM_C1) * (2.0f * sig12 + SSIM_C2);
    float den = (mu1 * mu1 + mu2 * mu2 + SSIM_C1) * (sig1 + sig2 + SSIM_C2);
    out[c] = num / den;
}

extern "C" void kernel_launch(void* const* d_in, const int* in_sizes, int n_in,
                              void* d_out, int out_size, void* d_ws, size_t ws_size,
                              hipStream_t stream) {
    const float* I1 = (const float*)d_in[0];
    const float* I2 = (const float*)d_in[1];
    float* out = (float*)d_out;
    float* ws  = (float*)d_ws;   // needs channels*SEG*5 floats (~30 KB)

    const int channels = out_size;                         // B*C = 96
    const long long total = (long long)in_sizes[0];        // 25,165,824
    const int n    = (int)(total / channels);              // 262,144 per channel
    const int nvec = n / 4;                                // float4 per channel
    const int iters = nvec / (SEG * TPB);                  // 16 for 512x512

    dim3 grid(SEG, channels);
    ssim_partial_kernel<<<grid, TPB, 0, stream>>>(I1, I2, ws, nvec, iters);

    const int fb = (channels + 127) / 128;
    ssim_final_kernel<<<fb, 128, 0, stream>>>(ws, out, channels, (float)n);
}